// XXX_Norm_8813272891444
// MI455X (gfx1250) — compile-verified
//
#include <hip/hip_runtime.h>
#include <hip/hip_bf16.h>

#define NTOT 409600
#define GSEG 1024
#define DFEAT 300
#define EPS_MAX 1e-12f
#define EPS_BN 1e-5f

typedef float vf2 __attribute__((ext_vector_type(2)));
typedef float vf4 __attribute__((ext_vector_type(4)));
typedef float vf8 __attribute__((ext_vector_type(8)));

__device__ __forceinline__ int lb_search(const int* __restrict__ s, int n, int val) {
    int lo = 0, hi = n;
    while (lo < hi) {
        int mid = (lo + hi) >> 1;
        if (s[mid] < val) lo = mid + 1; else hi = mid;
    }
    return lo;
}

// ---------------------------------------------------------------------------
// Pass 1: one block per segment (segment_ids are sorted -> contiguous runs).
// Computes per-segment column sums, column sums-of-squares, and scalar max|x|.
// Deterministic (fixed-order accumulation, LDS tree, no atomics).
// ---------------------------------------------------------------------------
__global__ __launch_bounds__(320) void seg_stats_kernel(
    const float* __restrict__ x, const int* __restrict__ seg,
    float* __restrict__ colsum, float* __restrict__ colsum2,
    float* __restrict__ gmax) {
    const int g = blockIdx.x;
    const int t = threadIdx.x;

    const int start = lb_search(seg, NTOT, g);
    const int end   = lb_search(seg, NTOT, g + 1);

    __shared__ vf4  red4[DFEAT];
    __shared__ float redm[320];

    vf4 sum = {0.f, 0.f, 0.f, 0.f};
    vf4 sq  = {0.f, 0.f, 0.f, 0.f};
    float vmax = 0.f;

    if (t < DFEAT) {
        const int j  = t % 75;   // float4 column group: cols 4j..4j+3
        const int r0 = t / 75;   // row phase 0..3
        #pragma unroll 4
        for (int n = start + r0; n < end; n += 4) {
            vf4 v = __builtin_nontemporal_load((const vf4*)(x + (size_t)n * DFEAT) + j);
            sum += v;
            sq  += v * v;
            float a0 = fmaxf(fabsf(v.x), fabsf(v.y));
            float a1 = fmaxf(fabsf(v.z), fabsf(v.w));
            vmax = fmaxf(vmax, fmaxf(a0, a1));
        }
    }

    if (t < DFEAT) red4[t] = sum;
    __syncthreads();
    if (t < 75) {
        vf4 s = red4[t] + red4[t + 75] + red4[t + 150] + red4[t + 225];
        *((vf4*)(colsum + (size_t)g * DFEAT) + t) = s;
    }
    __syncthreads();
    if (t < DFEAT) red4[t] = sq;
    __syncthreads();
    if (t < 75) {
        vf4 s = red4[t] + red4[t + 75] + red4[t + 150] + red4[t + 225];
        *((vf4*)(colsum2 + (size_t)g * DFEAT) + t) = s;
    }
    redm[t] = (t < DFEAT) ? vmax : 0.f;
    __syncthreads();
    if (t == 0) {
        float m = 0.f;
        for (int i = 0; i < 320; ++i) m = fmaxf(m, redm[i]);
        gmax[g] = m;
    }
}

// ---------------------------------------------------------------------------
// inv1[g] = 1/g_scale = rsqrt(max(eps, gmax)) ; inv2[g] = 1/g_scale^2
// ---------------------------------------------------------------------------
__global__ void inv_kernel(const float* __restrict__ gmax,
                           float* __restrict__ inv1, float* __restrict__ inv2) {
    int g = blockIdx.x * blockDim.x + threadIdx.x;
    if (g < GSEG) {
        float m = fmaxf(gmax[g], EPS_MAX);
        inv2[g] = 1.0f / m;
        inv1[g] = 1.0f / sqrtf(m);
    }
}

// ---------------------------------------------------------------------------
// Pass 2 (WMMA): sumx[d]  = sum_g colsum[g,d]  * inv1[g]
//                sumx2[d] = sum_g colsum2[g,d] * inv2[g]
// Mapped on V_WMMA_F32_16X16X4_F32:
//   A (16x4): M = feature (lane&15), K slots: VGPR0/1 x lane-half = g0..g0+3
//   B (4x16): same K slot layout, value inv[g0+k] replicated over all N columns
//   D (16x16): every column identical; lane 0 holds M=0..7, lane 16 M=8..15.
// ---------------------------------------------------------------------------
__global__ __launch_bounds__(32) void combine_wmma_kernel(
    const float* __restrict__ colsum, const float* __restrict__ colsum2,
    const float* __restrict__ inv1, const float* __restrict__ inv2,
    float* __restrict__ sumx, float* __restrict__ sumx2) {
    const int d0   = blockIdx.x * 16;
    const int lane = threadIdx.x;
    const int half = lane >> 4;   // 0: K=0,1   1: K=2,3
    const int l    = lane & 15;   // A row / B column
    const int dA   = d0 + l;
    const bool okA = (dA < DFEAT);

    vf8 accS = {0.f,0.f,0.f,0.f,0.f,0.f,0.f,0.f};
    vf8 accQ = {0.f,0.f,0.f,0.f,0.f,0.f,0.f,0.f};

    for (int g0 = 0; g0 < GSEG; g0 += 4) {
        const int k0 = g0 + 2 * half;
        vf2 A = {0.f, 0.f}, Q = {0.f, 0.f};
        if (okA) {
            A.x = colsum [(size_t)(k0    ) * DFEAT + dA];
            A.y = colsum [(size_t)(k0 + 1) * DFEAT + dA];
            Q.x = colsum2[(size_t)(k0    ) * DFEAT + dA];
            Q.y = colsum2[(size_t)(k0 + 1) * DFEAT + dA];
        }
        vf2 B1 = { inv1[k0], inv1[k0 + 1] };
        vf2 B2 = { inv2[k0], inv2[k0 + 1] };
        accS = __builtin_amdgcn_wmma_f32_16x16x4_f32(false, A, false, B1,
                                                     (short)0, accS, false, false);
        accQ = __builtin_amdgcn_wmma_f32_16x16x4_f32(false, Q, false, B2,
                                                     (short)0, accQ, false, false);
    }

    if (l == 0) {  // lanes 0 (M=0..7) and 16 (M=8..15), column N=0
        const int base = d0 + half * 8;
        #pragma unroll
        for (int r = 0; r < 8; ++r) {
            int d = base + r;
            if (d < DFEAT) { sumx[d] = accS[r]; sumx2[d] = accQ[r]; }
        }
    }
}

// ---------------------------------------------------------------------------
// Fold BN stats + weight/bias into per-feature affine: out = cA[d]*x + cB[d]
// ---------------------------------------------------------------------------
__global__ void finalize_kernel(const float* __restrict__ sumx,
                                const float* __restrict__ sumx2,
                                const float* __restrict__ w,
                                const float* __restrict__ b,
                                float* __restrict__ coefA,
                                float* __restrict__ coefB) {
    int d = blockIdx.x * blockDim.x + threadIdx.x;
    if (d < DFEAT) {
        float s = sumx[d], q = sumx2[d];
        float mean = s / (float)NTOT;
        float var  = (q - s * s / (float)NTOT) / (float)(NTOT - 1);  // ddof=1
        float rstd = 1.0f / sqrtf(var + EPS_BN);
        float a = w[d] * rstd;
        coefA[d] = a;
        coefB[d] = b[d] - mean * a;
    }
}

// ---------------------------------------------------------------------------
// Pass 3: out[n,d] = cA[d]*(x[n,d]*inv1[seg[n]]) + cB[d]
// float4 streaming with non-temporal load/store (read-once / write-once data).
// ---------------------------------------------------------------------------
#define ROWS_PER_BLOCK 64
__global__ __launch_bounds__(320) void apply_kernel(
    const float* __restrict__ x, const int* __restrict__ seg,
    const float* __restrict__ inv1, const float* __restrict__ coefA,
    const float* __restrict__ coefB, float* __restrict__ out) {
    const int t = threadIdx.x;
    if (t >= DFEAT) return;
    const int j  = t % 75;
    const int r0 = t / 75;

    const vf4 a  = *((const vf4*)coefA + j);
    const vf4 bb = *((const vf4*)coefB + j);

    const int nEnd = blockIdx.x * ROWS_PER_BLOCK + ROWS_PER_BLOCK;
    #pragma unroll 4
    for (int n = blockIdx.x * ROWS_PER_BLOCK + r0; n < nEnd; n += 4) {
        float s = inv1[seg[n]];
        vf4 v = __builtin_nontemporal_load((const vf4*)(x + (size_t)n * DFEAT) + j);
        vf4 o = a * (v * s) + bb;
        __builtin_nontemporal_store(o, (vf4*)(out + (size_t)n * DFEAT) + j);
    }
}

// ---------------------------------------------------------------------------
extern "C" void kernel_launch(void* const* d_in, const int* in_sizes, int n_in,
                              void* d_out, int out_size, void* d_ws, size_t ws_size,
                              hipStream_t stream) {
    const float* tensor = (const float*)d_in[0];
    const float* weight = (const float*)d_in[1];
    const float* bias   = (const float*)d_in[2];
    const int*   seg    = (const int*)d_in[3];
    float* out = (float*)d_out;

    float* ws      = (float*)d_ws;
    float* colsum  = ws;                         // 1024*300
    float* colsum2 = ws + 307200;                // 1024*300
    float* gmax    = ws + 614400;                // 1024
    float* inv1    = ws + 615424;                // 1024
    float* inv2    = ws + 616448;                // 1024
    float* sumx    = ws + 617472;                // 320
    float* sumx2   = ws + 617792;                // 320
    float* coefA   = ws + 618112;                // 320 (16B-aligned)
    float* coefB   = ws + 618432;                // 320 (16B-aligned)

    seg_stats_kernel<<<GSEG, 320, 0, stream>>>(tensor, seg, colsum, colsum2, gmax);
    inv_kernel<<<4, 256, 0, stream>>>(gmax, inv1, inv2);
    combine_wmma_kernel<<<(DFEAT + 15) / 16, 32, 0, stream>>>(colsum, colsum2,
                                                              inv1, inv2, sumx, sumx2);
    finalize_kernel<<<1, 320, 0, stream>>>(sumx, sumx2, weight, bias, coefA, coefB);
    apply_kernel<<<NTOT / ROWS_PER_BLOCK, 320, 0, stream>>>(tensor, seg, inv1,
                                                            coefA, coefB, out);
}